// GSplat_13417477833242
// MI455X (gfx1250) — compile-verified
//
#include <hip/hip_runtime.h>
#include <math.h>

// ---------------- problem constants ----------------
#define NREAL   714
#define NPAD    720          // 45 tiles * 16
#define NTILES  45
#define WIDTH   256
#define HEIGHT  256
#define FOCAL   128.0f       // 0.5*W/tan(pi/4)
#define CAM_T   8.0f
#define EPS2D   0.3f
#define A_MIN   (1.0f/255.0f)
#define A_MAX   0.999f

#define TAB_FLOATS (NPAD*12)   // 8640 floats: coef table (NPAD*8) then colOp (NPAD*4)

typedef __attribute__((ext_vector_type(2))) float v2f;
typedef __attribute__((ext_vector_type(8))) float v8f;
typedef __attribute__((ext_vector_type(4))) unsigned int v4u;
typedef __attribute__((ext_vector_type(4))) int v4i;
typedef __attribute__((ext_vector_type(8))) int v8i;

#if defined(__has_builtin)
#if __has_builtin(__builtin_amdgcn_tensor_load_to_lds)
#define HAVE_TDM 1
#endif
#endif

// ---------------- kernel 1: per-gaussian preprocessing ----------------
__global__ void gs_preprocess(const float* __restrict__ means,
                              const float* __restrict__ quats,
                              const float* __restrict__ scales,
                              const float* __restrict__ opacs,
                              const float* __restrict__ rgbs,
                              float* __restrict__ tz_out,
                              float* __restrict__ coef,
                              float* __restrict__ colOp)
{
    int i = blockIdx.x * blockDim.x + threadIdx.x;
    if (i >= NPAD) return;

    if (i >= NREAL) {
        #pragma unroll
        for (int k = 0; k < 8; ++k) coef[i*8 + k] = 0.0f;
        coef[i*8 + 5] = 1.0e8f;          // sigma=+big -> alpha=0
        colOp[i*4+0] = 0.0f; colOp[i*4+1] = 0.0f;
        colOp[i*4+2] = 0.0f; colOp[i*4+3] = 0.0f;
        tz_out[i] = 1.0e30f;
        return;
    }

    float qw = quats[i*4+0], qx = quats[i*4+1], qy = quats[i*4+2], qz = quats[i*4+3];
    float qinv = rsqrtf(qw*qw + qx*qx + qy*qy + qz*qz);
    qw *= qinv; qx *= qinv; qy *= qinv; qz *= qinv;

    float R00 = 1.0f - 2.0f*(qy*qy + qz*qz), R01 = 2.0f*(qx*qy - qw*qz), R02 = 2.0f*(qx*qz + qw*qy);
    float R10 = 2.0f*(qx*qy + qw*qz), R11 = 1.0f - 2.0f*(qx*qx + qz*qz), R12 = 2.0f*(qy*qz - qw*qx);
    float R20 = 2.0f*(qx*qz - qw*qy), R21 = 2.0f*(qy*qz + qw*qx), R22 = 1.0f - 2.0f*(qx*qx + qy*qy);

    float s0 = scales[i*3+0], s1 = scales[i*3+1], s2 = scales[i*3+2];
    float d0 = s0*s0, d1 = s1*s1, d2 = s2*s2;

    float V00 = R00*R00*d0 + R01*R01*d1 + R02*R02*d2;
    float V01 = R00*R10*d0 + R01*R11*d1 + R02*R12*d2;
    float V02 = R00*R20*d0 + R01*R21*d1 + R02*R22*d2;
    float V11 = R10*R10*d0 + R11*R11*d1 + R12*R12*d2;
    float V12 = R10*R20*d0 + R11*R21*d1 + R12*R22*d2;
    float V22 = R20*R20*d0 + R21*R21*d1 + R22*R22*d2;

    float tx = means[i*3+0], ty = means[i*3+1], tzv = means[i*3+2] + CAM_T;
    float invz = 1.0f / tzv;

    float J00 = FOCAL*invz, J02 = -FOCAL*tx*invz*invz;
    float J11 = FOCAL*invz, J12 = -FOCAL*ty*invz*invz;

    float JV00 = J00*V00 + J02*V02;
    float JV01 = J00*V01 + J02*V12;
    float JV02 = J00*V02 + J02*V22;
    float JV11 = J11*V11 + J12*V12;
    float JV12 = J11*V12 + J12*V22;

    float a = JV00*J00 + JV02*J02 + EPS2D;
    float b = JV01*J11 + JV02*J12;
    float c = JV11*J11 + JV12*J12 + EPS2D;

    float idet = 1.0f / (a*c - b*b);
    float cA = c*idet, cB = -b*idet, cC = a*idet;

    float mx = FOCAL*tx*invz + 0.5f*WIDTH;
    float my = FOCAL*ty*invz + 0.5f*HEIGHT;

    coef[i*8+0] = 0.5f*cA;
    coef[i*8+1] = 0.5f*cC;
    coef[i*8+2] = cB;
    coef[i*8+3] = -(cA*mx + cB*my);
    coef[i*8+4] = -(cC*my + cB*mx);
    coef[i*8+5] = 0.5f*cA*mx*mx + 0.5f*cC*my*my + cB*mx*my;
    coef[i*8+6] = 0.0f;
    coef[i*8+7] = 0.0f;

    colOp[i*4+0] = 1.0f/(1.0f + __expf(-rgbs[i*3+0]));
    colOp[i*4+1] = 1.0f/(1.0f + __expf(-rgbs[i*3+1]));
    colOp[i*4+2] = 1.0f/(1.0f + __expf(-rgbs[i*3+2]));
    colOp[i*4+3] = 1.0f/(1.0f + __expf(-opacs[i]));
    tz_out[i] = tzv;
}

// ---------------- kernel 2: O(N^2) rank sort + scatter ----------------
__global__ void gs_sort_scatter(const float* __restrict__ tz,
                                const float* __restrict__ coef,
                                const float* __restrict__ colOp,
                                float* __restrict__ sCoef,
                                float* __restrict__ sColOp)
{
    int i = blockIdx.x * blockDim.x + threadIdx.x;
    if (i >= NPAD) return;

    int rank;
    if (i >= NREAL) {
        rank = i;
    } else {
        float ti = tz[i];
        rank = 0;
        for (int j = 0; j < NREAL; ++j) {
            float tj = tz[j];
            rank += (tj < ti) || (tj == ti && j < i);
        }
    }
    #pragma unroll
    for (int k = 0; k < 8; ++k) sCoef[rank*8 + k] = coef[i*8 + k];
    #pragma unroll
    for (int k = 0; k < 4; ++k) sColOp[rank*4 + k] = colOp[i*4 + k];
}

// ---------------- kernel 3: WMMA tiled render with TDM-staged LDS ----------------
// gtab = sorted table: NPAD*8 coef floats immediately followed by NPAD*4 colOp floats.
__global__ void __launch_bounds__(256) gs_render(const float* __restrict__ gtab,
                                                 float* __restrict__ out)
{
    __shared__ float lds[TAB_FLOATS];   // 34560 bytes

#if defined(HAVE_TDM)
    if (threadIdx.x < 32) {
        // Tensor DMA descriptor: 1D copy of TAB_FLOATS f32 elements -> LDS.
        unsigned long long ga = (unsigned long long)(uintptr_t)gtab;
        unsigned int ldsa = (unsigned int)(uintptr_t)&lds[0];
        v4u g0;
        g0.x = 1u;                                        // count=1 (valid user D#)
        g0.y = ldsa;                                      // lds_addr
        g0.z = (unsigned int)(ga & 0xFFFFFFFFu);          // global_addr[31:0]
        g0.w = (unsigned int)((ga >> 32) & 0x01FFFFFFu)   // global_addr[56:32]
             | 0x80000000u;                               // type=2 ("image")
        v8i g1;
        g1[0] = 0x20000;                                  // data_size=2 (4 bytes), mask=0
        g1[1] = (int)((unsigned)TAB_FLOATS << 16);        // tensor_dim0[15:0] @ bits 63:48
        g1[2] = 0x00010000;                               // tensor_dim1 = 1
        g1[3] = (int)((unsigned)TAB_FLOATS << 16);        // tile_dim0 @ bits 127:112
        g1[4] = 0;                                        // tile_dim1/2 unused
        g1[5] = TAB_FLOATS;                               // tensor_dim0_stride low bits
        g1[6] = 0;
        g1[7] = 0;
        v4i gz = {0, 0, 0, 0};
#if __clang_major__ >= 23
        v8i gz8 = {0, 0, 0, 0, 0, 0, 0, 0};
        __builtin_amdgcn_tensor_load_to_lds(g0, g1, gz, gz, gz8, 0);
#else
        __builtin_amdgcn_tensor_load_to_lds(g0, g1, gz, gz, 0);
#endif
        __builtin_amdgcn_s_wait_tensorcnt(0);
    }
#else
    for (int k = threadIdx.x; k < TAB_FLOATS; k += 256) lds[k] = gtab[k];
#endif
    __syncthreads();

    int tid  = blockIdx.x * blockDim.x + threadIdx.x;
    int wave = tid >> 5;
    int lane = threadIdx.x & 31;
    bool hi  = lane >= 16;

    int p = wave * 16 + (lane & 15);
    float px = (float)(p & (WIDTH - 1)) + 0.5f;
    float py = (float)(p >> 8) + 0.5f;

    float m0 = px*px, m1 = py*py, m2 = px*py, m3 = px, m4 = py, m5 = 1.0f;

    v2f b1, b2;
    b1.x = hi ? m2 : m0;
    b1.y = hi ? m3 : m1;
    b2.x = hi ? 0.0f : m4;
    b2.y = hi ? 0.0f : m5;

    const float2* coefL = (const float2*)lds;               // 4 float2 per gaussian
    const float4* colL  = (const float4*)(lds + NPAD*8);    // rgb + opacity

    int aoff1 = hi ? 1 : 0;
    int aoff2 = hi ? 3 : 2;

    float T = 1.0f;
    float outR = 0.0f, outG = 0.0f, outB = 0.0f;

    #pragma unroll 3
    for (int tile = 0; tile < NTILES; ++tile) {
        int g = tile * 16 + (lane & 15);
        float2 a1f = coefL[g*4 + aoff1];
        float2 a2f = coefL[g*4 + aoff2];
        v2f a1; a1.x = a1f.x; a1.y = a1f.y;
        v2f a2; a2.x = a2f.x; a2.y = a2f.y;

        v8f acc = {};
        acc = __builtin_amdgcn_wmma_f32_16x16x4_f32(false, a1, false, b1,
                                                    (short)0, acc, false, false);
        acc = __builtin_amdgcn_wmma_f32_16x16x4_f32(false, a2, false, b2,
                                                    (short)0, acc, false, false);

        int gbase = tile * 16 + (hi ? 8 : 0);
        float tpart = 1.0f, cr = 0.0f, cg = 0.0f, cb = 0.0f;
        #pragma unroll
        for (int r = 0; r < 8; ++r) {
            float4 co = colL[gbase + r];
            float al = co.w * __expf(-acc[r]);
            al = fminf(al, A_MAX);
            al = (al < A_MIN) ? 0.0f : al;
            float w = tpart * al;
            cr += w * co.x;
            cg += w * co.y;
            cb += w * co.z;
            tpart *= (1.0f - al);
        }

        float t_o  = __shfl_xor(tpart, 16);
        float cr_o = __shfl_xor(cr, 16);
        float cg_o = __shfl_xor(cg, 16);
        float cb_o = __shfl_xor(cb, 16);

        float tLo  = hi ? t_o  : tpart;
        float tHi  = hi ? tpart : t_o;
        float crLo = hi ? cr_o : cr,  crHi = hi ? cr : cr_o;
        float cgLo = hi ? cg_o : cg,  cgHi = hi ? cg : cg_o;
        float cbLo = hi ? cb_o : cb,  cbHi = hi ? cb : cb_o;

        outR += T * (crLo + tLo * crHi);
        outG += T * (cgLo + tLo * cgHi);
        outB += T * (cbLo + tLo * cbHi);
        T *= tLo * tHi;
    }

    if (!hi) {
        out[p*3 + 0] = outR;
        out[p*3 + 1] = outG;
        out[p*3 + 2] = outB;
    }
}

// ---------------- launcher ----------------
extern "C" void kernel_launch(void* const* d_in, const int* in_sizes, int n_in,
                              void* d_out, int out_size, void* d_ws, size_t ws_size,
                              hipStream_t stream)
{
    // setup_inputs order: coords(unused), means, quats, scales, opacities, rgbs
    const float* means  = (const float*)d_in[1];
    const float* quats  = (const float*)d_in[2];
    const float* scales = (const float*)d_in[3];
    const float* opacs  = (const float*)d_in[4];
    const float* rgbs   = (const float*)d_in[5];

    float* ws     = (float*)d_ws;
    float* tz     = ws;                       // 720
    float* coef   = tz     + NPAD;            // 720*8
    float* colOp  = coef   + NPAD*8;          // 720*4
    float* sCoef  = colOp  + NPAD*4;          // 720*8   } contiguous 8640-float
    float* sColOp = sCoef  + NPAD*8;          // 720*4   } table staged by TDM

    gs_preprocess<<<dim3(3), dim3(256), 0, stream>>>(means, quats, scales, opacs, rgbs,
                                                     tz, coef, colOp);
    gs_sort_scatter<<<dim3(3), dim3(256), 0, stream>>>(tz, coef, colOp, sCoef, sColOp);

    gs_render<<<dim3(512), dim3(256), 0, stream>>>(sCoef, (float*)d_out);
}